// MultiHeadAttention_25202868093580
// MI455X (gfx1250) — compile-verified
//
#include <hip/hip_runtime.h>
#include <hip/hip_bf16.h>

typedef __attribute__((ext_vector_type(2))) float v2f;
typedef __attribute__((ext_vector_type(8))) float v8f;

#define Bsz 4
#define Dd 64
#define Nn 16384
#define Hh 8
#define Cc 16
#define Gg 32
#define G3c 32768
#define KVROWS 152 /* H*(C+3) */

// ---- gfx1250 async-to-LDS availability probe (compile-safe fallback) ----
#if defined(__has_builtin)
#  if __has_builtin(__builtin_amdgcn_global_load_async_to_lds_b32) && \
      __has_builtin(__builtin_amdgcn_s_wait_asynccnt)
#    define USE_ASYNC_LDS 1
#  endif
#endif
#ifndef USE_ASYNC_LDS
#  define USE_ASYNC_LDS 0
#endif

#if USE_ASYNC_LDS
typedef __attribute__((address_space(1))) int g_i32;   // global-AS int
typedef __attribute__((address_space(3))) int l_i32;   // LDS-AS int
#endif

// ---- workspace layout (in floats) ----
#define OFF_KV   0ull
#define SZ_KV    ((size_t)Bsz * KVROWS * Nn)     // 9,961,472
#define OFF_VAL  (OFF_KV + SZ_KV)
#define SZ_VAL   ((size_t)Bsz * Hh * Cc * Nn)    // 8,388,608
#define OFF_IDX  (OFF_VAL + SZ_VAL)
#define SZ_IDX   ((size_t)Bsz * Hh * 8 * Nn)     // 4,194,304
#define OFF_WT   (OFF_IDX + SZ_IDX)
#define SZ_WT    SZ_IDX
#define OFF_GRID (OFF_WT + SZ_WT)
#define SZ_GRID  ((size_t)Bsz * Hh * Cc * G3c)   // 16,777,216
#define OFF_CONV (OFF_GRID + SZ_GRID)
#define SZ_CONV  SZ_GRID

// ============================================================
// 1) kv = w_kv[152x64] @ x[b][64x16384] via V_WMMA_F32_16X16X4_F32
//    one wave per 16x16 output tile; 16 K-steps of 4.
//    A rows >=152 are clamped (their D rows are never stored),
//    so every load is unconditional -> no exec-mask diamonds.
// ============================================================
__global__ void __launch_bounds__(128)
kv_gemm_wmma(const float* __restrict__ x, const float* __restrict__ w_kv,
             float* __restrict__ kv) {
  int wid  = blockIdx.x * (blockDim.x >> 5) + (threadIdx.x >> 5);
  int lane = threadIdx.x & 31;
  int half = lane >> 4, lm = lane & 15;
  int nt   = wid & 1023;
  int rest = wid >> 10;
  int mt   = rest % 10;        // 10 M-tiles cover 152 rows
  int b    = rest / 10;
  int Mbase = mt * 16, Nbase = nt * 16;
  int arow  = Mbase + lm;
  if (arow >= KVROWS) arow = KVROWS - 1;   // clamp: result rows >=152 discarded
  const float* xb = x + (size_t)b * Dd * Nn;
  const v2f*   ar = (const v2f*)(w_kv + (size_t)arow * Dd);  // 8B-aligned pairs

  v8f acc = {};
#pragma unroll
  for (int kk = 0; kk < Dd; kk += 4) {
    v2f a = ar[(kk >> 1) + half];           // A[M=lm][kk+2*half +{0,1}]
    v2f bb;
    int ka = kk + 2 * half;
    bb.x = xb[(size_t)ka       * Nn + Nbase + lm];
    bb.y = xb[(size_t)(ka + 1) * Nn + Nbase + lm];
    acc = __builtin_amdgcn_wmma_f32_16x16x4_f32(false, a, false, bb,
                                                (short)0, acc, false, false);
  }
  float* kvb = kv + (size_t)b * KVROWS * Nn;
#pragma unroll
  for (int r = 0; r < 8; ++r) {
    int row = Mbase + r + half * 8;         // D layout: VGPR r = rows r / r+8
    if (row < KVROWS) kvb[(size_t)row * Nn + Nbase + lm] = acc[r];
  }
}

// ============================================================
// 2) prep: BN(keys) -> +orig_points -> per-head 3x3 +bias -> tanh
//    -> trilinear corner indices & weights; BN(values)
// ============================================================
__global__ void __launch_bounds__(256)
prep_kernel(const float* __restrict__ kv, const float* __restrict__ op,
            const float* __restrict__ kg, const float* __restrict__ kb,
            const float* __restrict__ km, const float* __restrict__ kvar,
            const float* __restrict__ vg, const float* __restrict__ vb,
            const float* __restrict__ vm, const float* __restrict__ vvar,
            const float* __restrict__ Wt, const float* __restrict__ bt,
            float* __restrict__ val, int* __restrict__ fidx,
            float* __restrict__ fw) {
  int id = blockIdx.x * blockDim.x + threadIdx.x;
  if (id >= Bsz * Hh * Nn) return;
  int n = id & (Nn - 1);
  int h = (id >> 14) & 7;
  int b = id >> 17;
  const float* kvb = kv + (size_t)b * KVROWS * Nn;

  float pt[3];
#pragma unroll
  for (int j = 0; j < 3; ++j) {
    int ch = h * 3 + j;
    float r = kvb[(size_t)ch * Nn + n];
    float v = (r - km[ch]) * (rsqrtf(kvar[ch] + 1e-5f) * kg[ch]) + kb[ch];
    pt[j] = op[((size_t)b * 3 + j) * Nn + n] + v;
  }
  int ib[3]; float fr[3];
#pragma unroll
  for (int i = 0; i < 3; ++i) {
    float t = bt[h * 3 + i];
#pragma unroll
    for (int j = 0; j < 3; ++j) t += Wt[(h * 3 + i) * 3 + j] * pt[j];
    float lat = tanhf(t);
    float pos = (lat + 1.0f) * (0.5f * (Gg - 1));
    float bs  = fminf(fmaxf(floorf(pos), 0.0f), (float)(Gg - 2));
    ib[i] = (int)bs;
    fr[i] = pos - bs;
  }
  size_t base8 = ((size_t)(b * Hh + h)) * 8 * Nn + n;
#pragma unroll
  for (int c8 = 0; c8 < 8; ++c8) {
    int oi = (c8 >> 2) & 1, oj = (c8 >> 1) & 1, ok = c8 & 1;
    int idx = ((ib[0] + oi) * Gg + ib[1] + oj) * Gg + ib[2] + ok;
    float w = (oi ? fr[0] : 1.0f - fr[0]) * (oj ? fr[1] : 1.0f - fr[1]) *
              (ok ? fr[2] : 1.0f - fr[2]);
    fidx[base8 + (size_t)c8 * Nn] = idx;
    fw  [base8 + (size_t)c8 * Nn] = w;
  }
  size_t vbase = ((size_t)(b * Hh + h)) * Cc * Nn + n;
#pragma unroll
  for (int c = 0; c < Cc; ++c) {
    int ch = h * Cc + c;                      // vbn channel
    float r = kvb[(size_t)(Hh * 3 + ch) * Nn + n];
    val[vbase + (size_t)c * Nn] =
        (r - vm[ch]) * (rsqrtf(vvar[ch] + 1e-5f) * vg[ch]) + vb[ch];
  }
}

// ============================================================
// 3) zero grid, then splat via native global_atomic_add_f32
// ============================================================
__global__ void zero_kernel(float* __restrict__ p, unsigned long long nfloat) {
  size_t i = (size_t)blockIdx.x * blockDim.x + threadIdx.x;
  size_t stride = (size_t)gridDim.x * blockDim.x;
  for (; i < nfloat; i += stride) p[i] = 0.0f;
}

__global__ void __launch_bounds__(256)
splat_kernel(const float* __restrict__ val, const int* __restrict__ fidx,
             const float* __restrict__ fw, float* __restrict__ grid) {
  int id = blockIdx.x * blockDim.x + threadIdx.x;
  if (id >= Bsz * Hh * Nn) return;
  int n = id & (Nn - 1);
  int h = (id >> 14) & 7;
  int b = id >> 17;
  size_t base8 = ((size_t)(b * Hh + h)) * 8 * Nn + n;
  int   idx[8]; float w[8];
#pragma unroll
  for (int c8 = 0; c8 < 8; ++c8) {
    idx[c8] = fidx[base8 + (size_t)c8 * Nn];
    w[c8]   = fw  [base8 + (size_t)c8 * Nn];
  }
  size_t vbase = ((size_t)(b * Hh + h)) * Cc * Nn + n;
  float* gb = grid + ((size_t)(b * Hh + h)) * Cc * G3c;
#pragma unroll
  for (int c = 0; c < Cc; ++c) {
    float v = val[vbase + (size_t)c * Nn];
    float* gc = gb + (size_t)c * G3c;
#pragma unroll
    for (int c8 = 0; c8 < 8; ++c8) {
      __hip_atomic_fetch_add(&gc[idx[c8]], v * w[c8],
                             __ATOMIC_RELAXED, __HIP_MEMORY_SCOPE_AGENT);
    }
  }
}

// ============================================================
// 4) grouped 3x3x3 conv as implicit GEMM:
//    per (b,h,d0,d1): out[16 c_out][32 d2] ; K = 16*27 = 432
//    A (weights) + halo input tile staged in LDS (async on gfx1250);
//    432-entry B-offset LUT kills the /27,/9,/3 VALU work in the loop.
// ============================================================
__global__ void __launch_bounds__(64)
conv3d_wmma(const float* __restrict__ grid, const float* __restrict__ conv_w,
            const float* __restrict__ conv_b, float* __restrict__ convo) {
  __shared__ float s_w[Cc * 432];          // 6912 floats = 27.0 KB
  __shared__ float s_in[Cc * 3 * 3 * 34];  // 4896 floats = 19.1 KB
  __shared__ int   s_bofs[432];            // B-row offset LUT
  int bid = blockIdx.x;
  int d1 = bid & 31, d0 = (bid >> 5) & 31, h = (bid >> 10) & 7, b = bid >> 13;
  int tid = threadIdx.x;

  const float* wsrc = conv_w + (size_t)h * (Cc * Cc * 27);
  for (int i = tid; i < Cc * 432; i += 64) s_w[i] = wsrc[i];

  // LUT: k = ci*27 + t0*9 + t1*3 + t2  ->  base offset into s_in (x added later)
  for (int k = tid; k < 432; k += 64) {
    int ci = k / 27, t = k - ci * 27;
    int t0 = t / 9; t -= t0 * 9; int t1 = t / 3; int t2 = t - t1 * 3;
    s_bofs[k] = ((ci * 3 + t0) * 3 + t1) * 34 + t2;
  }

  const float* gb = grid + ((size_t)(b * Hh + h) * Cc) * G3c;
  for (int i = tid; i < Cc * 306; i += 64) {
    int ci = i / 306, r = i - ci * 306;
    int zz = r / 102; r -= zz * 102;
    int yy = r / 34;  int xx = r - yy * 34;
    int gz = d0 + zz - 1, gy = d1 + yy - 1, gx = xx - 1;
    bool inr = (unsigned)gz < 32u && (unsigned)gy < 32u && (unsigned)gx < 32u;
#if USE_ASYNC_LDS
    if (inr) {
      const float* gp = &gb[(size_t)ci * G3c + (gz * Gg + gy) * Gg + gx];
      __builtin_amdgcn_global_load_async_to_lds_b32(
          (g_i32*)gp, (l_i32*)&s_in[i], 0, 0);
    } else {
      s_in[i] = 0.0f;
    }
#else
    float v = 0.0f;
    if (inr) v = gb[(size_t)ci * G3c + (gz * Gg + gy) * Gg + gx];
    s_in[i] = v;
#endif
  }
#if USE_ASYNC_LDS
  __builtin_amdgcn_s_wait_asynccnt(0);
#endif
  __syncthreads();

  int lane = tid & 31, half = lane >> 4, lm = lane & 15;
  int col = ((tid >> 5) << 4) + lm;        // wave0: d2 0..15, wave1: 16..31
  const float* aw = &s_w[lm * 432];
  v8f acc = {};
#pragma unroll 4
  for (int kk = 0; kk < 432; kk += 4) {
    int k0 = kk + 2 * half;
    v2f a = *(const v2f*)&aw[k0];           // 8B-aligned ds_load_b64
    v2f bb;
    bb.x = s_in[s_bofs[k0]     + col];
    bb.y = s_in[s_bofs[k0 + 1] + col];
    acc = __builtin_amdgcn_wmma_f32_16x16x4_f32(false, a, false, bb,
                                                (short)0, acc, false, false);
  }
  float* ob = convo + ((size_t)(b * Hh + h) * Cc) * G3c + (d0 * Gg + d1) * Gg + col;
#pragma unroll
  for (int r = 0; r < 8; ++r) {
    int m = r + half * 8;
    ob[(size_t)m * G3c] = acc[r] + conv_b[h * Cc + m];
  }
}

// ============================================================
// 5) slice (weighted gather over 8 corners) + BN + ReLU -> out
// ============================================================
__global__ void __launch_bounds__(256)
slice_kernel(const float* __restrict__ convo, const int* __restrict__ fidx,
             const float* __restrict__ fw, const float* __restrict__ ag,
             const float* __restrict__ ab, const float* __restrict__ am,
             const float* __restrict__ av, float* __restrict__ out) {
  int id = blockIdx.x * blockDim.x + threadIdx.x;
  if (id >= Bsz * Hh * Nn) return;
  int n = id & (Nn - 1);
  int h = (id >> 14) & 7;
  int b = id >> 17;
  size_t base8 = ((size_t)(b * Hh + h)) * 8 * Nn + n;
  int   idx[8]; float w[8];
#pragma unroll
  for (int c8 = 0; c8 < 8; ++c8) {
    idx[c8] = fidx[base8 + (size_t)c8 * Nn];
    w[c8]   = fw  [base8 + (size_t)c8 * Nn];
  }
  const float* cb = convo + ((size_t)(b * Hh + h)) * Cc * G3c;
#pragma unroll
  for (int c = 0; c < Cc; ++c) {
    const float* cc = cb + (size_t)c * G3c;
    float s = 0.0f;
#pragma unroll
    for (int c8 = 0; c8 < 8; ++c8) s += cc[idx[c8]] * w[c8];
    int ch = h * Cc + c;
    float o = (s - am[ch]) * (rsqrtf(av[ch] + 1e-5f) * ag[ch]) + ab[ch];
    out[((size_t)b * (Hh * Cc) + ch) * Nn + n] = fmaxf(o, 0.0f);
  }
}

// ============================================================
extern "C" void kernel_launch(void* const* d_in, const int* in_sizes, int n_in,
                              void* d_out, int out_size, void* d_ws, size_t ws_size,
                              hipStream_t stream) {
  const float* x    = (const float*)d_in[0];
  const float* op   = (const float*)d_in[1];
  const float* wkv  = (const float*)d_in[2];
  const float* kg   = (const float*)d_in[3];
  const float* kb   = (const float*)d_in[4];
  const float* km   = (const float*)d_in[5];
  const float* kvar = (const float*)d_in[6];
  const float* vg   = (const float*)d_in[7];
  const float* vb   = (const float*)d_in[8];
  const float* vm   = (const float*)d_in[9];
  const float* vvar = (const float*)d_in[10];
  const float* Wt   = (const float*)d_in[11];
  const float* bt   = (const float*)d_in[12];
  const float* cw   = (const float*)d_in[13];
  const float* cb   = (const float*)d_in[14];
  const float* ag   = (const float*)d_in[15];
  const float* abt  = (const float*)d_in[16];
  const float* am   = (const float*)d_in[17];
  const float* av   = (const float*)d_in[18];

  float* ws    = (float*)d_ws;
  float* kv    = ws + OFF_KV;
  float* val   = ws + OFF_VAL;
  int*   fidx  = (int*)(ws + OFF_IDX);
  float* fw    = ws + OFF_WT;
  float* grid  = ws + OFF_GRID;
  float* convo = ws + OFF_CONV;
  float* out   = (float*)d_out;

  // 1) GEMM: 4 batches x 10 M-tiles x 1024 N-tiles = 40960 waves / 4 per block
  kv_gemm_wmma<<<10240, 128, 0, stream>>>(x, wkv, kv);
  // 2) per-point prep (B*H*N = 524288 threads)
  prep_kernel<<<2048, 256, 0, stream>>>(kv, op, kg, kb, km, kvar,
                                        vg, vb, vm, vvar, Wt, bt,
                                        val, fidx, fw);
  // 3) zero + splat
  zero_kernel<<<8192, 256, 0, stream>>>(grid, (unsigned long long)SZ_GRID);
  splat_kernel<<<2048, 256, 0, stream>>>(val, fidx, fw, grid);
  // 4) grouped conv: one block per (b,h,d0,d1) = 32768 blocks of 2 waves
  conv3d_wmma<<<32768, 64, 0, stream>>>(grid, cw, cb, convo);
  // 5) slice + BN + ReLU
  slice_kernel<<<2048, 256, 0, stream>>>(convo, fidx, fw, ag, abt, am, av, out);
}